// ShifthWiseConv2dImplicit_78477642433092
// MI455X (gfx1250) — compile-verified
//
#include <hip/hip_runtime.h>
#include <hip/hip_bf16.h>

typedef float v2f __attribute__((ext_vector_type(2)));
typedef float v8f __attribute__((ext_vector_type(8)));

#define REP_N   196
#define GHOST_N 60
#define IN_C    256
#define BATCH   32
#define HW      56
#define HWSQ    (HW*HW)
#define LDSH    80    // logical rows -6..73
#define LDSW    84    // logical cols -6..77, stride 84 -> conflict-free Toeplitz column reads
#define NPIX    (32.0f*56.0f*56.0f)
#define STATS_F (6*REP_N)
#define PARAM_OFF STATS_F
#define CP      48    // padded Toeplitz stride: index (t+15), t in [-15,30], band t in [0,14]

// ---------------------------------------------------------------------------
// Fused 16x16 tile: vertical 15x3 conv (banded-Toeplitz A x input-panel B) and
// horizontal 3x15 conv (input-panel A x banded-Toeplitz B), interleaved so the
// two WMMA accumulation chains are independent (latency hiding within a wave).
//   accV[m,n] += sum_k cVp[p][k-m+15] * X[y0-6+k, x0+n+p-1]
//   accH[m,n] += sum_k X[y0+m+p-1, x0-6+k] * cHp[p][k-n+15]
// cVp/cHp are [3][48] zero-padded bands -> unconditional LDS loads, no EXEC math.
// ---------------------------------------------------------------------------
__device__ __forceinline__ void conv_tile_VH(const float* __restrict__ cVp,
                                             const float* __restrict__ cHp,
                                             const float* __restrict__ Xl,
                                             int y0, int x0, v8f& accV, v8f& accH) {
    const int lane = threadIdx.x & 31;
    const int mn = lane & 15;          // M for A-side, N for B-side
    const int hi = lane >> 4;          // K half select
#pragma unroll
    for (int ks = 0; ks < 8; ++ks) {
        const int k0 = ks * 4 + hi * 2;
        const int t0 = k0 - mn + 15;   // padded Toeplitz index, in [0,45]
#pragma unroll
        for (int p = 0; p < 3; ++p) {
            // ---- vertical: A = Toeplitz coeffs, B = input columns ----
            v2f av, bv;
            av.x = cVp[p * CP + t0];
            av.y = cVp[p * CP + t0 + 1];
            const int vrow = y0 + k0;               // lds row for K=k0
            const int vcol = x0 + mn + p + 5;       // image col x0+n+(p-1)
            bv.x = Xl[vrow * LDSW + vcol];
            bv.y = Xl[(vrow + 1) * LDSW + vcol];
            accV = __builtin_amdgcn_wmma_f32_16x16x4_f32(false, av, false, bv,
                                                         (short)0, accV, false, false);
            // ---- horizontal: A = input rows, B = Toeplitz coeffs ----
            v2f ah, bh;
            const int hrow = y0 + mn + p + 5;       // image row y0+m+(p-1)
            const int hcol = x0 + k0;               // image col x0-6+k0
            ah.x = Xl[hrow * LDSW + hcol];
            ah.y = Xl[hrow * LDSW + hcol + 1];
            bh.x = cHp[p * CP + t0];
            bh.y = cHp[p * CP + t0 + 1];
            accH = __builtin_amdgcn_wmma_f32_16x16x4_f32(false, ah, false, bh,
                                                         (short)0, accH, false, false);
        }
    }
}

__device__ __forceinline__ void load_channel_lds(float* Xl, const float* __restrict__ xin,
                                                 int tid) {
    for (int idx = tid; idx < LDSH * LDSW; idx += 512) {
        const int r = idx / LDSW, cc = idx % LDSW;
        const int ir = r - 6, ic = cc - 6;
        float v = 0.0f;
        if ((unsigned)ir < (unsigned)HW && (unsigned)ic < (unsigned)HW)
            v = xin[ir * HW + ic];
        Xl[idx] = v;
    }
}

// ===========================================================================
// Pass 1: per-channel sum / sumsq of lora1 (15x3), lora2 (3x15), small (3x3)
// ===========================================================================
__global__ __launch_bounds__(512) void stats_kernel(const float* __restrict__ in,
                                                    const float* __restrict__ w1,
                                                    const float* __restrict__ w2,
                                                    const int* __restrict__ rep_idx,
                                                    float* __restrict__ ws) {
    __shared__ float Xl[LDSH * LDSW];
    __shared__ float cVp[3 * CP], cHp[3 * CP], cS[9];
    __shared__ float sacc[6];
    const int c = blockIdx.x, b = blockIdx.y, tid = threadIdx.x;
    const int ic = rep_idx[c];
    const float* xin = in + ((size_t)b * IN_C + ic) * HWSQ;

    if (tid < 3 * CP) {                     // vertical band: t=3i+kh, p=kw
        const int p = tid / CP, t = tid % CP - 15;
        float v = 0.0f;
        if (t >= 0 && t <= 14) {
            const int i = t / 3, kh = t % 3;
            const int o = (c * 5 + i) * 9 + kh * 3 + p;
            v = w1[o] + w2[o];
        }
        cVp[tid] = v;
    } else if (tid < 6 * CP) {              // horizontal band: p=kh, t=3i+kw
        const int idx = tid - 3 * CP;
        const int p = idx / CP, t = idx % CP - 15;
        float v = 0.0f;
        if (t >= 0 && t <= 14) {
            const int i = t / 3, kw = t % 3;
            const int o = (c * 5 + i) * 9 + p * 3 + kw;
            v = w1[o] + w2[o];
        }
        cHp[idx] = v;
    } else if (tid < 6 * CP + 9) {          // small 3x3: replica i=2
        const int k = tid - 6 * CP, o = (c * 5 + 2) * 9 + k;
        cS[k] = w1[o] + w2[o];
    } else if (tid < 6 * CP + 15) {
        sacc[tid - 6 * CP - 9] = 0.0f;
    }
    load_channel_lds(Xl, xin, tid);
    __syncthreads();

    const int wv = tid >> 5;
    const int y0 = (wv >> 2) * 16, x0 = (wv & 3) * 16;
    v8f accV = {}, accH = {};
    conv_tile_VH(cVp, cHp, Xl, y0, x0, accV, accH);

    const int lane = tid & 31, n = lane & 15, hi = lane >> 4;
    float s1 = 0, q1 = 0, s2 = 0, q2 = 0, s3 = 0, q3 = 0;
#pragma unroll
    for (int j = 0; j < 8; ++j) {
        const int h = y0 + j + 8 * hi, w = x0 + n;
        if (h < HW && w < HW) {
            const float v1 = accV[j], v2 = accH[j];
            float sm = 0.0f;
#pragma unroll
            for (int kh = 0; kh < 3; ++kh)
#pragma unroll
                for (int kw = 0; kw < 3; ++kw)
                    sm = fmaf(cS[kh * 3 + kw], Xl[(h + kh + 5) * LDSW + (w + kw + 5)], sm);
            s1 += v1; q1 = fmaf(v1, v1, q1);
            s2 += v2; q2 = fmaf(v2, v2, q2);
            s3 += sm; q3 = fmaf(sm, sm, q3);
        }
    }
    atomicAdd(&sacc[0], s1); atomicAdd(&sacc[1], q1);
    atomicAdd(&sacc[2], s2); atomicAdd(&sacc[3], q2);
    atomicAdd(&sacc[4], s3); atomicAdd(&sacc[5], q3);
    __syncthreads();
    if (tid < 6) atomicAdd(&ws[tid * REP_N + c], sacc[tid]);
}

// ===========================================================================
// Pass 2: fold BN into per-channel affine params (a1,a2,a3,const)
// ===========================================================================
__global__ __launch_bounds__(256) void finalize_kernel(const float* __restrict__ g1, const float* __restrict__ b1,
                                                       const float* __restrict__ g2, const float* __restrict__ b2,
                                                       const float* __restrict__ g3, const float* __restrict__ b3,
                                                       float* __restrict__ ws) {
    const int c = blockIdx.x * blockDim.x + threadIdx.x;
    if (c >= REP_N) return;
    const float m1 = ws[0 * REP_N + c] / NPIX, v1 = fmaxf(ws[1 * REP_N + c] / NPIX - m1 * m1, 0.0f);
    const float m2 = ws[2 * REP_N + c] / NPIX, v2 = fmaxf(ws[3 * REP_N + c] / NPIX - m2 * m2, 0.0f);
    const float m3 = ws[4 * REP_N + c] / NPIX, v3 = fmaxf(ws[5 * REP_N + c] / NPIX - m3 * m3, 0.0f);
    const float a1 = g1[c] * rsqrtf(v1 + 1e-5f);
    const float a2 = g2[c] * rsqrtf(v2 + 1e-5f);
    const float a3 = g3[c] * rsqrtf(v3 + 1e-5f);
    float* p = ws + PARAM_OFF + 4 * c;
    p[0] = a1; p[1] = a2; p[2] = a3;
    p[3] = b1[c] + b2[c] + b3[c] - a1 * m1 - a2 * m2 - a3 * m3;
}

// ===========================================================================
// Pass 3: y = a1*conv15x3 + a3*conv3x3 (folded into cVp) + a2*conv3x15 + x + const
// ===========================================================================
__global__ __launch_bounds__(512) void main_kernel(const float* __restrict__ in,
                                                   const float* __restrict__ w1,
                                                   const float* __restrict__ w2,
                                                   const int* __restrict__ rep_idx,
                                                   const float* __restrict__ ws,
                                                   float* __restrict__ out) {
    __shared__ float Xl[LDSH * LDSW];
    __shared__ float cVp[3 * CP], cHp[3 * CP];
    const int c = blockIdx.x, b = blockIdx.y, tid = threadIdx.x;
    const float* prm = ws + PARAM_OFF + 4 * c;
    const float a1 = prm[0], a2 = prm[1], a3 = prm[2], cc = prm[3];
    const int ic = rep_idx[c];
    const float* xin = in + ((size_t)b * IN_C + ic) * HWSQ;

    if (tid < 3 * CP) {
        const int p = tid / CP, t = tid % CP - 15;
        float v = 0.0f;
        if (t >= 0 && t <= 14) {
            const int i = t / 3, kh = t % 3;
            const int o = (c * 5 + i) * 9 + kh * 3 + p;
            v = a1 * (w1[o] + w2[o]);
            if (t >= 5 && t <= 7) {         // fold a3 * small taps (dy in {-1,0,1})
                const int os = (c * 5 + 2) * 9 + (t - 5) * 3 + p;
                v = fmaf(a3, w1[os] + w2[os], v);
            }
        }
        cVp[tid] = v;
    } else if (tid < 6 * CP) {
        const int idx = tid - 3 * CP;
        const int p = idx / CP, t = idx % CP - 15;
        float v = 0.0f;
        if (t >= 0 && t <= 14) {
            const int i = t / 3, kw = t % 3;
            const int o = (c * 5 + i) * 9 + p * 3 + kw;
            v = a2 * (w1[o] + w2[o]);
        }
        cHp[idx] = v;
    }
    load_channel_lds(Xl, xin, tid);
    __syncthreads();

    const int wv = tid >> 5;
    const int y0 = (wv >> 2) * 16, x0 = (wv & 3) * 16;
    v8f accV = {}, accH = {};
    conv_tile_VH(cVp, cHp, Xl, y0, x0, accV, accH);

    const int lane = tid & 31, n = lane & 15, hi = lane >> 4;
    float* outp = out + ((size_t)b * IN_C + c) * HWSQ;   // rep channels occupy 0..195
#pragma unroll
    for (int j = 0; j < 8; ++j) {
        const int h = y0 + j + 8 * hi, w = x0 + n;
        if (h < HW && w < HW)
            outp[h * HW + w] = accV[j] + accH[j] + Xl[(h + 6) * LDSW + (w + 6)] + cc;
    }
}

// ===========================================================================
// Ghost channels: pure copy into out channels 196..255
// ===========================================================================
__global__ __launch_bounds__(256) void ghost_kernel(const float* __restrict__ in,
                                                    const int* __restrict__ ghost_idx,
                                                    float* __restrict__ out) {
    const int total = BATCH * GHOST_N * HWSQ;
    const int tid = blockIdx.x * blockDim.x + threadIdx.x;
    if (tid >= total) return;
    const int b = tid / (GHOST_N * HWSQ);
    const int r = tid % (GHOST_N * HWSQ);
    const int j = r / HWSQ, pix = r % HWSQ;
    out[((size_t)b * IN_C + (REP_N + j)) * HWSQ + pix] =
        in[((size_t)b * IN_C + ghost_idx[j]) * HWSQ + pix];
}

extern "C" void kernel_launch(void* const* d_in, const int* in_sizes, int n_in,
                              void* d_out, int out_size, void* d_ws, size_t ws_size,
                              hipStream_t stream) {
    const float* x  = (const float*)d_in[0];
    const float* w1 = (const float*)d_in[1];
    const float* w2 = (const float*)d_in[2];
    const float* g1 = (const float*)d_in[3];
    const float* b1 = (const float*)d_in[4];
    const float* g2 = (const float*)d_in[5];
    const float* b2 = (const float*)d_in[6];
    const float* g3 = (const float*)d_in[7];
    const float* b3 = (const float*)d_in[8];
    const int* ghost_idx = (const int*)d_in[9];
    const int* rep_idx   = (const int*)d_in[10];
    float* out = (float*)d_out;
    float* ws  = (float*)d_ws;

    hipMemsetAsync(ws, 0, STATS_F * sizeof(float), stream);

    dim3 grid(REP_N, BATCH);
    stats_kernel<<<grid, 512, 0, stream>>>(x, w1, w2, rep_idx, ws);
    finalize_kernel<<<1, 256, 0, stream>>>(g1, b1, g2, b2, g3, b3, ws);
    main_kernel<<<grid, 512, 0, stream>>>(x, w1, w2, rep_idx, ws, out);

    const int total = BATCH * GHOST_N * HWSQ;
    ghost_kernel<<<(total + 255) / 256, 256, 0, stream>>>(x, ghost_idx, out);
}